// _PointnetSAModuleBase_9242769621774
// MI455X (gfx1250) — compile-verified
//
#include <hip/hip_runtime.h>

typedef __attribute__((ext_vector_type(2))) float v2f;
typedef __attribute__((ext_vector_type(8))) float v8f;

#define B_   16
#define N_   8192
#define S_   1024
#define K_   32
#define CIN  35
#define C1   64
#define C3   128
#define RAD2 0.04f
#define NROWS (B_ * S_ * K_)   // 524288
#define BN_EPS_ 1e-5f

// ---------------------------------------------------------------------------
// workspace layout (bytes):
//   [0,   2048)      : 512 floats of BN stats: sum1[64] sq1[64] sum2[64] sq2[64] sum3[128] sq3[128]
//   [2048,67584)     : fps_idx   int[16*1024]
//   [67584, ~2.16MB) : ball idx  int[16*1024*32]
// ---------------------------------------------------------------------------

__global__ void zero_stats_kernel(float* __restrict__ stats) {
    stats[threadIdx.x] = 0.0f;   // launched with 512 threads
}

// ---------------------------------------------------------------------------
// Furthest point sampling: one block per batch, 512 threads, 16 pts/thread.
// Emits fps_idx and new_xyz (first output chunk of d_out).
// ---------------------------------------------------------------------------
__global__ __launch_bounds__(512) void fps_kernel(const float* __restrict__ xyz,
                                                  float* __restrict__ new_xyz,
                                                  int* __restrict__ fps_idx) {
    const int b    = blockIdx.x;
    const int tid  = threadIdx.x;
    const int lane = tid & 31;
    const int wid  = tid >> 5;
    const float* xb = xyz + (size_t)b * N_ * 3;

    float md[16];
#pragma unroll
    for (int i = 0; i < 16; ++i) md[i] = 1e10f;

    __shared__ float swv[16];
    __shared__ int   swi[16];
    __shared__ int   sfar;

    int far = 0;
    for (int t = 0; t < S_; ++t) {
        if (tid == 0) {
            fps_idx[b * S_ + t] = far;
            new_xyz[(b * S_ + t) * 3 + 0] = xb[far * 3 + 0];
            new_xyz[(b * S_ + t) * 3 + 1] = xb[far * 3 + 1];
            new_xyz[(b * S_ + t) * 3 + 2] = xb[far * 3 + 2];
        }
        const float cx = xb[far * 3 + 0];
        const float cy = xb[far * 3 + 1];
        const float cz = xb[far * 3 + 2];

        float bv = -1.0f;
        int   bi = 0;
#pragma unroll
        for (int i = 0; i < 16; ++i) {
            const int p = i * 512 + tid;
            const float dx = xb[p * 3 + 0] - cx;
            const float dy = xb[p * 3 + 1] - cy;
            const float dz = xb[p * 3 + 2] - cz;
            const float d = dx * dx + dy * dy + dz * dz;
            md[i] = fminf(md[i], d);
            if (md[i] > bv || (md[i] == bv && p < bi)) { bv = md[i]; bi = p; }
        }
#pragma unroll
        for (int off = 16; off > 0; off >>= 1) {
            const float ov = __shfl_down(bv, off);
            const int   oi = __shfl_down(bi, off);
            if (ov > bv || (ov == bv && oi < bi)) { bv = ov; bi = oi; }
        }
        if (lane == 0) { swv[wid] = bv; swi[wid] = bi; }
        __syncthreads();
        if (tid == 0) {
            float xv = swv[0]; int xi = swi[0];
            for (int w = 1; w < 16; ++w)
                if (swv[w] > xv || (swv[w] == xv && swi[w] < xi)) { xv = swv[w]; xi = swi[w]; }
            sfar = xi;
        }
        __syncthreads();
        far = sfar;
    }
}

// ---------------------------------------------------------------------------
// Ball query: one wave (32 lanes) per query point. Ordered append via ballot
// + prefix popcount reproduces "first 32 in-radius indices, pad with first".
// ---------------------------------------------------------------------------
__global__ __launch_bounds__(256) void ballquery_kernel(const float* __restrict__ xyz,
                                                        const float* __restrict__ new_xyz,
                                                        const int* __restrict__ fps_idx,
                                                        int* __restrict__ idx_out) {
    __shared__ int sidx[8][K_];
    const int w    = threadIdx.x >> 5;
    const int lane = threadIdx.x & 31;
    const int q    = blockIdx.x * 8 + w;          // 0..16383
    const int b    = q >> 10;
    const float* xb = xyz + (size_t)b * N_ * 3;
    const float nx = new_xyz[q * 3 + 0];
    const float ny = new_xyz[q * 3 + 1];
    const float nz = new_xyz[q * 3 + 2];
    const unsigned lmask = (1u << lane) - 1u;

    int count = 0;
    for (int chunk = 0; chunk < (N_ / 32) && count < K_; ++chunk) {
        const int j = chunk * 32 + lane;
        const float dx = xb[j * 3 + 0] - nx;
        const float dy = xb[j * 3 + 1] - ny;
        const float dz = xb[j * 3 + 2] - nz;
        const bool pred = (dx * dx + dy * dy + dz * dz) < RAD2;
        const unsigned mask = (unsigned)__ballot(pred);
        const int pos = count + __popc(mask & lmask);
        if (pred && pos < K_) sidx[w][pos] = j;
        count += __popc(mask);
    }
    if (count > K_) count = K_;
    __syncthreads();
    int v = (lane < count) ? sidx[w][lane] : 0;
    int first = __shfl(v, 0);
    if (count == 0) first = fps_idx[q];   // safety: query point itself is always in-radius
    if (lane >= count) v = first;
    idx_out[q * K_ + lane] = v;
}

// ---------------------------------------------------------------------------
// Fused MLP chain with V_WMMA_F32_16X16X4_F32.
// PASS 1: y1 stats.  PASS 2: y2 stats.  PASS 3: y3 stats.  PASS 4: output.
// Block = 128 threads = 4 waves; 2 groups/block; wave handles a 16-row half.
// ---------------------------------------------------------------------------
__device__ __forceinline__ void accum_stats(const v8f* acc, int ntiles,
                                            float* sS, float* sS2, int lane) {
    const int l16 = lane & 15;
    for (int nt = 0; nt < ntiles; ++nt) {
        float sA = 0.0f, sQ = 0.0f;
#pragma unroll
        for (int r = 0; r < 8; ++r) { const float v = acc[nt][r]; sA += v; sQ += v * v; }
        sA += __shfl_xor(sA, 16);
        sQ += __shfl_xor(sQ, 16);
        if (lane < 16) {
            atomicAdd(&sS[nt * 16 + l16], sA);
            atomicAdd(&sS2[nt * 16 + l16], sQ);
        }
    }
}

template <int PASS>
__global__ __launch_bounds__(128) void mlp_kernel(
    const float* __restrict__ xyz, const float* __restrict__ feat,
    const float* __restrict__ W1, const float* __restrict__ g1, const float* __restrict__ b1,
    const float* __restrict__ W2, const float* __restrict__ g2, const float* __restrict__ b2,
    const float* __restrict__ W3, const float* __restrict__ g3, const float* __restrict__ b3,
    const float* __restrict__ new_xyz, const int* __restrict__ idx,
    float* __restrict__ stats, float* __restrict__ out_feat) {

    __shared__ float tile[4][16][68];       // per-wave activation tile (padded stride)
    __shared__ float sS[128], sS2[128];     // block-level stats partials
    __shared__ float gmax[2][2][128];       // per-group half-tile column maxima

    const int tid   = threadIdx.x;
    const int wid   = tid >> 5;
    const int lane  = tid & 31;
    const int grp   = wid >> 1;
    const int half  = wid & 1;
    const int group = blockIdx.x * 2 + grp; // b*1024 + s
    const int b     = group >> 10;
    const int s     = group & 1023;
    const int l16   = lane & 15;
    const int hh    = lane >> 4;            // 0 or 1
    const float invN = 1.0f / (float)NROWS;

    if (PASS != 4 && tid < 128) { sS[tid] = 0.0f; sS2[tid] = 0.0f; }

    __builtin_prefetch(W1, 0, 1);
    __builtin_prefetch(W2, 0, 1);
    __builtin_prefetch(W3, 0, 1);

    // ---- stage gathered input tile (16 rows x 36 channels, ch35 = 0 pad) ----
    {
        const int row = l16;
        const int k   = half * 16 + row;
        const int j   = idx[group * K_ + k];
        const int ch0 = hh * 18;
        const float nx0 = new_xyz[group * 3 + 0];
        const float nx1 = new_xyz[group * 3 + 1];
        const float nx2 = new_xyz[group * 3 + 2];
        for (int c = ch0; c < ch0 + 18; ++c) {
            float v;
            if (c < 3)        v = xyz[((size_t)b * N_ + j) * 3 + c] - (c == 0 ? nx0 : (c == 1 ? nx1 : nx2));
            else if (c < CIN) v = feat[((size_t)b * 32 + (c - 3)) * N_ + j];
            else              v = 0.0f;
            tile[wid][row][c] = v;
        }
    }
    __syncthreads();

    v8f acc[8];

    // ---------------- layer 1: (16x36) @ (36x64) ----------------
#pragma unroll
    for (int nt = 0; nt < 4; ++nt)
#pragma unroll
        for (int e = 0; e < 8; ++e) acc[nt][e] = 0.0f;

#pragma unroll
    for (int kc = 0; kc < 9; ++kc) {
        const int kb = kc * 4 + hh * 2;
        v2f a; a[0] = tile[wid][l16][kb]; a[1] = tile[wid][l16][kb + 1];
#pragma unroll
        for (int nt = 0; nt < 4; ++nt) {
            const int n = nt * 16 + l16;
            v2f w;
            w[0] = (kb     < CIN) ? W1[n * CIN + kb]     : 0.0f;
            w[1] = (kb + 1 < CIN) ? W1[n * CIN + kb + 1] : 0.0f;
            acc[nt] = __builtin_amdgcn_wmma_f32_16x16x4_f32(false, a, false, w,
                                                            (short)0, acc[nt], false, false);
        }
    }

    if (PASS == 1) {
        accum_stats(acc, 4, sS, sS2, lane);
        __syncthreads();
        if (tid < C1) { atomicAdd(&stats[0 + tid], sS[tid]); atomicAdd(&stats[64 + tid], sS2[tid]); }
        return;
    }

    // ---- BN1 + ReLU, transpose through LDS ----
#pragma unroll
    for (int nt = 0; nt < 4; ++nt) {
        const int n = nt * 16 + l16;
        const float mean = stats[0 + n] * invN;
        const float var  = stats[64 + n] * invN - mean * mean;
        const float sc   = rsqrtf(var + BN_EPS_) * g1[n];
        const float sh   = b1[n] - mean * sc;
#pragma unroll
        for (int r = 0; r < 8; ++r) acc[nt][r] = fmaxf(acc[nt][r] * sc + sh, 0.0f);
    }
    __syncthreads();
#pragma unroll
    for (int nt = 0; nt < 4; ++nt) {
        const int n = nt * 16 + l16;
#pragma unroll
        for (int r = 0; r < 8; ++r) tile[wid][r + hh * 8][n] = acc[nt][r];
    }
    __syncthreads();

    // ---------------- layer 2: (16x64) @ (64x64) ----------------
#pragma unroll
    for (int nt = 0; nt < 4; ++nt)
#pragma unroll
        for (int e = 0; e < 8; ++e) acc[nt][e] = 0.0f;

#pragma unroll
    for (int kc = 0; kc < 16; ++kc) {
        const int kb = kc * 4 + hh * 2;
        v2f a; a[0] = tile[wid][l16][kb]; a[1] = tile[wid][l16][kb + 1];
#pragma unroll
        for (int nt = 0; nt < 4; ++nt) {
            const int n = nt * 16 + l16;
            v2f w; w[0] = W2[n * 64 + kb]; w[1] = W2[n * 64 + kb + 1];
            acc[nt] = __builtin_amdgcn_wmma_f32_16x16x4_f32(false, a, false, w,
                                                            (short)0, acc[nt], false, false);
        }
    }

    if (PASS == 2) {
        accum_stats(acc, 4, sS, sS2, lane);
        __syncthreads();
        if (tid < C1) { atomicAdd(&stats[128 + tid], sS[tid]); atomicAdd(&stats[192 + tid], sS2[tid]); }
        return;
    }

    // ---- BN2 + ReLU, transpose through LDS ----
#pragma unroll
    for (int nt = 0; nt < 4; ++nt) {
        const int n = nt * 16 + l16;
        const float mean = stats[128 + n] * invN;
        const float var  = stats[192 + n] * invN - mean * mean;
        const float sc   = rsqrtf(var + BN_EPS_) * g2[n];
        const float sh   = b2[n] - mean * sc;
#pragma unroll
        for (int r = 0; r < 8; ++r) acc[nt][r] = fmaxf(acc[nt][r] * sc + sh, 0.0f);
    }
    __syncthreads();
#pragma unroll
    for (int nt = 0; nt < 4; ++nt) {
        const int n = nt * 16 + l16;
#pragma unroll
        for (int r = 0; r < 8; ++r) tile[wid][r + hh * 8][n] = acc[nt][r];
    }
    __syncthreads();

    // ---------------- layer 3: (16x64) @ (64x128) ----------------
#pragma unroll
    for (int nt = 0; nt < 8; ++nt)
#pragma unroll
        for (int e = 0; e < 8; ++e) acc[nt][e] = 0.0f;

#pragma unroll
    for (int kc = 0; kc < 16; ++kc) {
        const int kb = kc * 4 + hh * 2;
        v2f a; a[0] = tile[wid][l16][kb]; a[1] = tile[wid][l16][kb + 1];
#pragma unroll
        for (int nt = 0; nt < 8; ++nt) {
            const int n = nt * 16 + l16;
            v2f w; w[0] = W3[n * 64 + kb]; w[1] = W3[n * 64 + kb + 1];
            acc[nt] = __builtin_amdgcn_wmma_f32_16x16x4_f32(false, a, false, w,
                                                            (short)0, acc[nt], false, false);
        }
    }

    if (PASS == 3) {
        accum_stats(acc, 8, sS, sS2, lane);
        __syncthreads();
        if (tid < C3) { atomicAdd(&stats[256 + tid], sS[tid]); atomicAdd(&stats[384 + tid], sS2[tid]); }
        return;
    }

    // ---- PASS 4: BN3 + ReLU + max over 32 samples, write (B,128,S) ----
#pragma unroll
    for (int nt = 0; nt < 8; ++nt) {
        const int n = nt * 16 + l16;
        const float mean = stats[256 + n] * invN;
        const float var  = stats[384 + n] * invN - mean * mean;
        const float sc   = rsqrtf(var + BN_EPS_) * g3[n];
        const float sh   = b3[n] - mean * sc;
        float m = 0.0f;   // all post-ReLU values are >= 0
#pragma unroll
        for (int r = 0; r < 8; ++r) m = fmaxf(m, fmaxf(acc[nt][r] * sc + sh, 0.0f));
        m = fmaxf(m, __shfl_xor(m, 16));
        if (lane < 16) gmax[grp][half][n] = m;
    }
    __syncthreads();
    if (half == 0 && lane < 16) {
#pragma unroll
        for (int nt = 0; nt < 8; ++nt) {
            const int c = nt * 16 + lane;
            const float v = fmaxf(gmax[grp][0][c], gmax[grp][1][c]);
            out_feat[((size_t)b * C3 + c) * S_ + s] = v;
        }
    }
}

// ---------------------------------------------------------------------------
extern "C" void kernel_launch(void* const* d_in, const int* in_sizes, int n_in,
                              void* d_out, int out_size, void* d_ws, size_t ws_size,
                              hipStream_t stream) {
    const float* xyz  = (const float*)d_in[0];
    const float* feat = (const float*)d_in[1];
    const float* W1   = (const float*)d_in[2];
    const float* g1   = (const float*)d_in[3];
    const float* b1   = (const float*)d_in[4];
    const float* W2   = (const float*)d_in[5];
    const float* g2   = (const float*)d_in[6];
    const float* b2   = (const float*)d_in[7];
    const float* W3   = (const float*)d_in[8];
    const float* g3   = (const float*)d_in[9];
    const float* b3   = (const float*)d_in[10];

    float* new_xyz  = (float*)d_out;                 // 16*1024*3
    float* out_feat = (float*)d_out + B_ * S_ * 3;   // 16*128*1024

    float* stats   = (float*)d_ws;                                 // 512 floats
    int*   fpsidx  = (int*)((char*)d_ws + 2048);                   // 16384 ints
    int*   ballidx = (int*)((char*)d_ws + 2048 + B_ * S_ * 4);     // 524288 ints

    zero_stats_kernel<<<1, 512, 0, stream>>>(stats);
    fps_kernel<<<B_, 512, 0, stream>>>(xyz, new_xyz, fpsidx);
    ballquery_kernel<<<(B_ * S_) / 8, 256, 0, stream>>>(xyz, new_xyz, fpsidx, ballidx);

    const dim3 g(B_ * S_ / 2);
    mlp_kernel<1><<<g, 128, 0, stream>>>(xyz, feat, W1, g1, b1, W2, g2, b2, W3, g3, b3,
                                         new_xyz, ballidx, stats, out_feat);
    mlp_kernel<2><<<g, 128, 0, stream>>>(xyz, feat, W1, g1, b1, W2, g2, b2, W3, g3, b3,
                                         new_xyz, ballidx, stats, out_feat);
    mlp_kernel<3><<<g, 128, 0, stream>>>(xyz, feat, W1, g1, b1, W2, g2, b2, W3, g3, b3,
                                         new_xyz, ballidx, stats, out_feat);
    mlp_kernel<4><<<g, 128, 0, stream>>>(xyz, feat, W1, g1, b1, W2, g2, b2, W3, g3, b3,
                                         new_xyz, ballidx, stats, out_feat);
}